// MEANLayer_58652073394402
// MI455X (gfx1250) — compile-verified
//
#include <hip/hip_runtime.h>

typedef __attribute__((ext_vector_type(2))) float v2f;
typedef __attribute__((ext_vector_type(8))) float v8f;

#define DD 32
#define EPSV 1e-5f

// ---------------- kernel 0: zero out/stat buffers ----------------
__global__ void k_zero(float* out, float* nodeCnt, float* sum, float* sumsq,
                       int* counts, int N, int R) {
  int stride = gridDim.x * blockDim.x;
  int total = N * DD + N + 64 + R;
  for (int i = blockIdx.x * blockDim.x + threadIdx.x; i < total; i += stride) {
    if (i < N * DD)               out[i] = 0.0f;
    else if (i < N * DD + N)      nodeCnt[i - N * DD] = 0.0f;
    else if (i < N * DD + N + 32) sum[i - N * DD - N] = 0.0f;
    else if (i < N * DD + N + 64) sumsq[i - N * DD - N - 32] = 0.0f;
    else                          counts[i - N * DD - N - 64] = 0;
  }
}

// ---------------- kernel 1: relation histogram + node in-degree ----------------
__global__ void k_hist(const int* __restrict__ etype, const int* __restrict__ dst,
                       int* counts, float* nodeCnt, int E) {
  int stride = gridDim.x * blockDim.x;
  for (int i = blockIdx.x * blockDim.x + threadIdx.x; i < E; i += stride) {
    atomicAdd(&counts[etype[i]], 1);
    atomicAdd(&nodeCnt[dst[i]], 1.0f);
  }
}

// ---------------- kernel 2: serial prefix sums (R=100, trivial) ----------------
__global__ void k_prefix(const int* counts, int* offsets, int* cursor,
                         int* tilePref, int* numTiles, int R) {
  if (blockIdx.x == 0 && threadIdx.x == 0) {
    int acc = 0, tacc = 0;
    for (int r = 0; r < R; ++r) {
      offsets[r] = acc; cursor[r] = acc; tilePref[r] = tacc;
      acc += counts[r];
      tacc += (counts[r] + 15) >> 4;
    }
    offsets[R] = acc; tilePref[R] = tacc; *numTiles = tacc;
  }
}

// ---------------- kernel 3a: tile table (tiles never cross a relation) ----------------
__global__ void k_tiles(const int* offsets, const int* tilePref,
                        int* tileRel, int* tileStart, int R) {
  int r = blockIdx.x;
  if (r >= R) return;
  int s = offsets[r];
  int nt = (offsets[r + 1] - s + 15) >> 4;
  int base = tilePref[r];
  for (int i = threadIdx.x; i < nt; i += blockDim.x) {
    tileRel[base + i] = r;
    tileStart[base + i] = s + 16 * i;
  }
}

// ---------------- kernel 3b: scatter edge ids into relation-sorted order ----------------
__global__ void k_scatter(const int* __restrict__ etype, int* cursor, int* perm, int E) {
  int stride = gridDim.x * blockDim.x;
  for (int i = blockIdx.x * blockDim.x + threadIdx.x; i < E; i += stride) {
    int p = atomicAdd(&cursor[etype[i]], 1);
    perm[p] = i;
  }
}

// ---------------- kernel 4: per-tile GEMM via V_WMMA_F32_16X16X4_F32 ----------------
// One wave = 16 edges (same relation) x 32 output channels.
// A (16x32 f32): lane holds row M=lane&15; VGPR pair per k-step holds
//   K = 4*kk + 2*(lane>>4) + {0,1}   (ISA 32-bit A 16x4 layout)
// B (32x32 f32, K-major): lane holds col N=lane&15 (+16 for second C tile),
//   same K split across lane halves (mirrors A layout).
// C/D (16x16 f32, 8 VGPRs): VGPR i, lanes 0-15 -> M=i, lanes 16-31 -> M=8+i.
__global__ void __launch_bounds__(256)
k_wmma(const float* __restrict__ h, const float* __restrict__ W,
       const int* __restrict__ src, const int* __restrict__ perm,
       const int* __restrict__ offsets,
       const int* __restrict__ tileRel, const int* __restrict__ tileStart,
       const int* __restrict__ numTiles,
       float* __restrict__ msg, float* sum, float* sumsq) {
  __shared__ float bsum[DD];
  __shared__ float bss[DD];
  int tid = threadIdx.x;
  if (tid < DD) { bsum[tid] = 0.0f; bss[tid] = 0.0f; }
  __syncthreads();

  int lane = tid & 31;
  int gw = blockIdx.x * (blockDim.x >> 5) + (tid >> 5);   // wave-uniform
  bool active = gw < *numTiles;                           // wave-uniform branch,
  if (active) {                                           // EXEC all-1s inside (WMMA req)
    int r     = tileRel[gw];
    int start = tileStart[gw];
    int valid = offsets[r + 1] - start;
    if (valid > 16) valid = 16;

    int row  = lane & 15;
    int half = lane >> 4;
    int arow = (row < valid) ? row : (valid - 1);         // clamp pad rows (masked later)
    int eA   = perm[start + arow];
    const float* __restrict__ hrow = h + (long)src[eA] * DD;
    const float* __restrict__ Wr   = W + (long)r * DD * DD;

    v8f c0 = {0.f, 0.f, 0.f, 0.f, 0.f, 0.f, 0.f, 0.f};
    v8f c1 = c0;
#pragma unroll
    for (int kk = 0; kk < 8; ++kk) {
      int k = 4 * kk + 2 * half;
      v2f a, b0, b1;
      a.x  = hrow[k];
      a.y  = hrow[k + 1];
      b0.x = Wr[k * DD + row];           b0.y = Wr[(k + 1) * DD + row];
      b1.x = Wr[k * DD + 16 + row];      b1.y = Wr[(k + 1) * DD + 16 + row];
      c0 = __builtin_amdgcn_wmma_f32_16x16x4_f32(false, a, false, b0,
                                                 (short)0, c0, false, false);
      c1 = __builtin_amdgcn_wmma_f32_16x16x4_f32(false, a, false, b1,
                                                 (short)0, c1, false, false);
    }

    // ReLU, store msg (original edge-id order), accumulate BN stats.
    float ls0 = 0.f, lq0 = 0.f, ls1 = 0.f, lq1 = 0.f;
#pragma unroll
    for (int i = 0; i < 8; ++i) {
      int M = i + 8 * half;
      if (M < valid) {
        int eM = perm[start + M];
        float v0 = c0[i] > 0.f ? c0[i] : 0.f;
        float v1 = c1[i] > 0.f ? c1[i] : 0.f;
        msg[(long)eM * DD + row]      = v0;
        msg[(long)eM * DD + 16 + row] = v1;
        ls0 += v0; lq0 += v0 * v0;
        ls1 += v1; lq1 += v1 * v1;
      }
    }
    atomicAdd(&bsum[row], ls0);      atomicAdd(&bss[row], lq0);
    atomicAdd(&bsum[16 + row], ls1); atomicAdd(&bss[16 + row], lq1);
  }
  __syncthreads();
  if (tid < DD) {
    atomicAdd(&sum[tid], bsum[tid]);
    atomicAdd(&sumsq[tid], bss[tid]);
  }
}

// ---------------- kernel 5: finalize BN statistics ----------------
__global__ void k_stats(const float* sum, const float* sumsq,
                        float* mu, float* rsig, int E) {
  int c = threadIdx.x;
  if (c < DD) {
    float m = sum[c] / (float)E;
    float v = sumsq[c] / (float)E - m * m;   // biased variance
    mu[c] = m;
    rsig[c] = rsqrtf(v + EPSV);
  }
}

// ---------------- kernel 6: normalize + scatter-add into destination nodes ----------------
__global__ void k_norm_scatter(const float* __restrict__ msg, const int* __restrict__ dst,
                               const float* __restrict__ mu, const float* __restrict__ rsig,
                               const float* __restrict__ gamma, const float* __restrict__ beta,
                               float* out, int E) {
  long stride = (long)gridDim.x * blockDim.x;
  long total = (long)E * DD;
  for (long i = (long)blockIdx.x * blockDim.x + threadIdx.x; i < total; i += stride) {
    int e = (int)(i >> 5);
    int c = (int)(i & 31);
    float v = (msg[i] - mu[c]) * rsig[c] * gamma[c] + beta[c];
    atomicAdd(&out[(long)dst[e] * DD + c], v);
  }
}

// ---------------- kernel 7: divide by degree (segment mean) ----------------
__global__ void k_div(float* out, const float* __restrict__ nodeCnt, int N) {
  int i = blockIdx.x * blockDim.x + threadIdx.x;
  if (i < N * DD) {
    float c = nodeCnt[i >> 5];
    out[i] /= (c > 1.0f ? c : 1.0f);
  }
}

extern "C" void kernel_launch(void* const* d_in, const int* in_sizes, int n_in,
                              void* d_out, int out_size, void* d_ws, size_t ws_size,
                              hipStream_t stream) {
  (void)n_in; (void)out_size; (void)ws_size;
  const float* h     = (const float*)d_in[0];
  const float* W     = (const float*)d_in[1];
  const float* gamma = (const float*)d_in[2];
  const float* beta  = (const float*)d_in[3];
  const int*   src   = (const int*)d_in[4];
  const int*   dst   = (const int*)d_in[5];
  const int*   etype = (const int*)d_in[6];
  float* out = (float*)d_out;

  int N = in_sizes[0] / DD;
  int R = in_sizes[1] / (DD * DD);
  int E = in_sizes[4];
  int maxTiles = (E + 15) / 16 + R;

  // workspace carve (256B aligned fields)
  char* p = (char*)d_ws;
  auto carve = [&](size_t bytes) { char* q = p; p += (bytes + 255) & ~(size_t)255; return q; };
  float* msg      = (float*)carve((size_t)E * DD * sizeof(float));
  int*   perm     = (int*)  carve((size_t)E * sizeof(int));
  int*   tileRel  = (int*)  carve((size_t)maxTiles * sizeof(int));
  int*   tileStart= (int*)  carve((size_t)maxTiles * sizeof(int));
  int*   counts   = (int*)  carve((size_t)R * sizeof(int));
  int*   offsets  = (int*)  carve((size_t)(R + 1) * sizeof(int));
  int*   cursor   = (int*)  carve((size_t)R * sizeof(int));
  int*   tilePref = (int*)  carve((size_t)(R + 1) * sizeof(int));
  int*   numTiles = (int*)  carve(sizeof(int));
  float* sum      = (float*)carve(DD * sizeof(float));
  float* sumsq    = (float*)carve(DD * sizeof(float));
  float* mu       = (float*)carve(DD * sizeof(float));
  float* rsig     = (float*)carve(DD * sizeof(float));
  float* nodeCnt  = (float*)carve((size_t)N * sizeof(float));

  k_zero<<<1024, 256, 0, stream>>>(out, nodeCnt, sum, sumsq, counts, N, R);
  k_hist<<<1024, 256, 0, stream>>>(etype, dst, counts, nodeCnt, E);
  k_prefix<<<1, 32, 0, stream>>>(counts, offsets, cursor, tilePref, numTiles, R);
  k_tiles<<<R, 64, 0, stream>>>(offsets, tilePref, tileRel, tileStart, R);
  k_scatter<<<1024, 256, 0, stream>>>(etype, cursor, perm, E);
  int wmmaBlocks = (maxTiles + 7) / 8;   // 8 waves / block, 1 tile / wave
  k_wmma<<<wmmaBlocks, 256, 0, stream>>>(h, W, src, perm, offsets,
                                         tileRel, tileStart, numTiles,
                                         msg, sum, sumsq);
  k_stats<<<1, 32, 0, stream>>>(sum, sumsq, mu, rsig, E);
  k_norm_scatter<<<4096, 256, 0, stream>>>(msg, dst, mu, rsig, gamma, beta, out, E);
  k_div<<<(N * DD + 255) / 256, 256, 0, stream>>>(out, nodeCnt, N);
}